// TopKGating_17746804867277
// MI455X (gfx1250) — compile-verified
//
#include <hip/hip_runtime.h>
#include <math.h>

typedef float v2f __attribute__((ext_vector_type(2)));
typedef float v4f __attribute__((ext_vector_type(4)));
typedef float v8f __attribute__((ext_vector_type(8)));

#define B_TOK 4096
#define S_TOK 197
#define D_TOK 768
#define E_EXP 16
#define NOISE_EPS 0.01f
#define INV_SQRT2 0.70710678118654752f

// ---------------------------------------------------------------------------
// Kernel 1: mean over S.  tokens (B,S,D) -> pooled (B,D).
// Pure HBM streaming: 2.48 GB read @ 23.3 TB/s is the runtime floor (~107us).
// 192 threads * v4f = 768 floats per row; nontemporal loads (data is
// touched exactly once and is far larger than the 192MB L2).
// ---------------------------------------------------------------------------
__global__ __launch_bounds__(192) void pool_kernel(const float* __restrict__ tokens,
                                                   float* __restrict__ pooled) {
  const int b = blockIdx.x;
  const int t = threadIdx.x;                       // 0..191, 4 floats each
  const float* base = tokens + (size_t)b * (size_t)(S_TOK * D_TOK) + (size_t)t * 4;
  v4f acc = {0.0f, 0.0f, 0.0f, 0.0f};
#pragma unroll 4
  for (int s = 0; s < S_TOK; ++s) {
    v4f v = __builtin_nontemporal_load((const v4f*)(base + (size_t)s * D_TOK));
    acc += v;
  }
  const float inv = 1.0f / (float)S_TOK;
  acc *= inv;
  *(v4f*)(pooled + (size_t)b * D_TOK + (size_t)t * 4) = acc;
}

// ---------------------------------------------------------------------------
// Kernel 2: per 16-row tile: dual fp32 WMMA GEMM (clean logits + noise
// logits), then gating epilogue. One wave (32 lanes) per block.
//   A (16x4 f32): lane m / m+16 holds row m; VGPR0,1 = K {0,1} / {2,3}
//   B (4x16 f32): lanes 0-15 = N; VGPR0,1 = K {0,1} lanes 0-15, {2,3} hi
//   C (16x16 f32): VGPR r, lanes 0-15 -> M=r, lanes 16-31 -> M=r+8
// ---------------------------------------------------------------------------
__global__ __launch_bounds__(32) void gate_kernel(const float* __restrict__ pooled,
                                                  const float* __restrict__ noise,
                                                  const float* __restrict__ w_gate,
                                                  const float* __restrict__ w_noise,
                                                  float* __restrict__ out_logits,
                                                  float* __restrict__ out_sel,
                                                  float* __restrict__ out_ew,
                                                  float* __restrict__ ws_load) {
  __shared__ float sCg[16][16];
  __shared__ float sCn[16][16];
  __shared__ float sP[16][16];

  const int lane = threadIdx.x;
  const int half = lane >> 4;      // 0: K {0,1}, 1: K {2,3}
  const int m    = lane & 15;      // A: row   B: column (expert)
  const int row0 = blockIdx.x * 16;

  v8f cg = {0.0f, 0.0f, 0.0f, 0.0f, 0.0f, 0.0f, 0.0f, 0.0f};
  v8f cn = {0.0f, 0.0f, 0.0f, 0.0f, 0.0f, 0.0f, 0.0f, 0.0f};

  const float* arow = pooled + (size_t)(row0 + m) * D_TOK + 2 * half;

  for (int k = 0; k < D_TOK; k += 4) {
    v2f a = *(const v2f*)(arow + k);               // pooled[row][k+2h .. +1]
    const int kb = k + 2 * half;
    v2f bg, bn;
    bg.x = w_gate [(size_t)kb       * E_EXP + m];
    bg.y = w_gate [(size_t)(kb + 1) * E_EXP + m];
    bn.x = w_noise[(size_t)kb       * E_EXP + m];
    bn.y = w_noise[(size_t)(kb + 1) * E_EXP + m];
    cg = __builtin_amdgcn_wmma_f32_16x16x4_f32(false, a, false, bg, (short)0, cg, false, false);
    cn = __builtin_amdgcn_wmma_f32_16x16x4_f32(false, a, false, bn, (short)0, cn, false, false);
  }

  // spill C tiles to LDS so one lane can see a whole row
#pragma unroll
  for (int r = 0; r < 8; ++r) {
    sCg[r + 8 * half][m] = cg[r];
    sCn[r + 8 * half][m] = cn[r];
  }
  __syncthreads();

  float clean[16], sd[16], rl[16];
  if (lane < 16) {
    const int brow = row0 + lane;
#pragma unroll
    for (int e = 0; e < 16; ++e) {
      clean[e] = sCg[lane][e];
      float x  = sCn[lane][e];
      // softplus(x) = max(x,0) + log1p(exp(-|x|))  (stable)
      float sp = fmaxf(x, 0.0f) + log1pf(expf(-fabsf(x)));
      sd[e] = sp + NOISE_EPS;
      rl[e] = clean[e] + noise[(size_t)brow * E_EXP + e] * sd[e];
      out_logits[(size_t)brow * E_EXP + e] = rl[e];
    }

    // top-3 (strict > keeps lowest index on ties, matching lax.top_k)
    float v0 = -3.402823e38f, v1 = -3.402823e38f, v2 = -3.402823e38f;
    int i0 = 0, i1 = 0;
#pragma unroll
    for (int e = 0; e < 16; ++e) {
      float v = rl[e];
      if (v > v0)      { v2 = v1; v1 = v0; i1 = i0; v0 = v; i0 = e; }
      else if (v > v1) { v2 = v1; v1 = v; i1 = e; }
      else if (v > v2) { v2 = v; }
    }
    out_sel[(size_t)brow * 2 + 0] = (float)i0;
    out_sel[(size_t)brow * 2 + 1] = (float)i1;

    // softmax over the top-2
    float t  = expf(v1 - v0);
    float w0 = 1.0f / (1.0f + t);
    float w1 = t * w0;

#pragma unroll
    for (int e = 0; e < 16; ++e) {
      float ew = (e == i0) ? w0 : ((e == i1) ? w1 : 0.0f);
      out_ew[(size_t)brow * E_EXP + e] = ew;
      // load probability: threshold v2 (3rd) if in top-2 region, else v1 (2nd)
      float thr = (rl[e] > v2) ? v2 : v1;
      float z   = (clean[e] - thr) / sd[e];
      sP[lane][e] = 0.5f * (1.0f + erff(z * INV_SQRT2));
    }
  }
  __syncthreads();

  // deterministic per-block partial of `load` (fixed summation order)
  if (lane < 16) {
    float s = 0.0f;
#pragma unroll
    for (int mm = 0; mm < 16; ++mm) s += sP[mm][lane];
    ws_load[(size_t)blockIdx.x * E_EXP + lane] = s;
  }
}

// ---------------------------------------------------------------------------
// Kernel 3: deterministic final reduction of the 256 per-block partials.
// ---------------------------------------------------------------------------
__global__ __launch_bounds__(32) void load_reduce_kernel(const float* __restrict__ ws_load,
                                                         float* __restrict__ out_load,
                                                         int nblocks) {
  const int e = threadIdx.x;
  if (e < E_EXP) {
    float s = 0.0f;
    for (int b = 0; b < nblocks; ++b) s += ws_load[(size_t)b * E_EXP + e];
    out_load[e] = s;
  }
}

extern "C" void kernel_launch(void* const* d_in, const int* in_sizes, int n_in,
                              void* d_out, int out_size, void* d_ws, size_t ws_size,
                              hipStream_t stream) {
  const float* tokens  = (const float*)d_in[0];   // (4096,197,768)
  const float* noise   = (const float*)d_in[1];   // (4096,16)
  const float* w_gate  = (const float*)d_in[2];   // (768,16)
  const float* w_noise = (const float*)d_in[3];   // (768,16)

  float* out = (float*)d_out;
  float* out_logits = out;                                   // 4096*16
  float* out_sel    = out_logits + (size_t)B_TOK * E_EXP;    // 4096*2
  float* out_ew     = out_sel    + (size_t)B_TOK * 2;        // 4096*16
  float* out_load   = out_ew     + (size_t)B_TOK * E_EXP;    // 16

  float* pooled  = (float*)d_ws;                             // 4096*768 f32
  float* ws_load = pooled + (size_t)B_TOK * D_TOK;           // 256*16 f32

  const int ntiles = B_TOK / 16;                             // 256

  pool_kernel<<<B_TOK, 192, 0, stream>>>(tokens, pooled);
  gate_kernel<<<ntiles, 32, 0, stream>>>(pooled, noise, w_gate, w_noise,
                                         out_logits, out_sel, out_ew, ws_load);
  load_reduce_kernel<<<1, 32, 0, stream>>>(ws_load, out_load, ntiles);
}